// DynamicGaussianBlur_76579266887900
// MI455X (gfx1250) — compile-verified
//
#include <hip/hip_runtime.h>
#include <cstdint>

typedef __attribute__((ext_vector_type(2))) float v2f;
typedef __attribute__((ext_vector_type(8))) float v8f;

#define RAD        12
#define KW         25           // kernel support
#define TILE_T     40           // 16 outputs + 2*12 halo taps
#define TILE_PITCH 42           // padded pitch: conflict-free, even (float2-aligned)
#define WPAD_BASE  16           // wpad[16+j] = w[j], zeros elsewhere (j in [-15,40])

// ---------------------------------------------------------------------------
// CDNA5 async memory->LDS copy (GLOBAL_LOAD_ASYNC_TO_LDS_B32, ASYNCcnt).
// lds_off is the wave-relative LDS byte address = low 32 bits of the generic
// shared pointer (flat->LDS mapping truncates to addr[31:0]).
// ---------------------------------------------------------------------------
__device__ __forceinline__ void async_ld_f32(uint32_t lds_off, const float* gaddr) {
  asm volatile("global_load_async_to_lds_b32 %0, %1, off"
               :: "v"(lds_off), "v"(gaddr) : "memory");
}
__device__ __forceinline__ void wait_async0() {
  asm volatile("s_wait_asynccnt 0x0" ::: "memory");
}

// ---------------------------------------------------------------------------
// Per-(sample,axis) Gaussian weights, matching the reference:
//   rad = max(1, floor(4*sigma)); w = exp(-0.5*off^2/sigma^2) * (|off|<=rad)
//   normalized; delta kernel when sigma <= 0.01.
// ---------------------------------------------------------------------------
__global__ void gauss_weights_kernel(const float* __restrict__ sigma,
                                     float* __restrict__ wout) {
  int id = threadIdx.x;
  if (id >= 12) return;
  float s  = sigma[id];                       // sigma[b*3+ax]
  float sc = fmaxf(s, 1e-6f);
  float rad = fmaxf(floorf(4.0f * s), 1.0f);
  float w[KW];
  float sum = 0.0f;
  for (int k = 0; k < KW; ++k) {
    float off = (float)(k - RAD);
    float v = __expf(-0.5f * off * off / (sc * sc));
    v = (fabsf(off) <= rad) ? v : 0.0f;
    w[k] = v;
    sum += v;
  }
  float inv = 1.0f / sum;
  for (int k = 0; k < KW; ++k) {
    float v = w[k] * inv;
    if (s <= 0.01f) v = (k == RAD) ? 1.0f : 0.0f;   // delta kernel
    wout[id * KW + k] = v;
  }
}

// ---------------------------------------------------------------------------
// One 1-D blur pass along an arbitrary axis, one wave32 per 16x16 output tile.
// out[16x16] = Toeplitz(16x40) * X(40x16) via 10 chained v_wmma_f32_16x16x4_f32.
//   BLUR_ON_N=1 (W pass, Sa==1): blur axis on N (Toeplitz in B) -> stores
//     lane-contiguous along W; staging t-major (contiguous global reads).
//   BLUR_ON_N=0 (H/D pass, Sl==1): blur axis on M (Toeplitz in A) -> lines on
//     the lane dimension along contiguous W; staging n-major.
// ---------------------------------------------------------------------------
template <int BLUR_ON_N>
__global__ __launch_bounds__(32)
void blur_axis_wmma(const float* __restrict__ in, float* __restrict__ out,
                    const float* __restrict__ wAll, int waxis,
                    int Sa, int La,            // blur-axis stride / extent
                    int Sl,                    // line-axis stride (extent mult of 16)
                    int So,                    // outer-axis stride (blockIdx.y)
                    int Sbc,                   // per-(b,c) stride (blockIdx.z)
                    int tilesL) {
  __shared__ float tile[16 * TILE_PITCH];     // tile[line][t], t = 0..39
  __shared__ float wpad[64];                  // zero-padded weights

  const int lane = threadIdx.x;               // wave32: 0..31, EXEC all ones
  const int ta   = blockIdx.x / tilesL;
  const int tl   = blockIdx.x - ta * tilesL;
  const int base = blockIdx.z * Sbc + blockIdx.y * So;
  const int b    = blockIdx.z >> 1;           // blockIdx.z = b*2 + c

  const int a0 = ta * 16;
  const int l0 = tl * 16;

  // ---- stage 40x16 input tile straight into LDS via async copies
  //      (replicate-clamped along the blur axis); 640 f32 = 20 iters/lane.
  if (BLUR_ON_N) {
    // Sa == 1: consecutive lanes -> consecutive global addresses (t-major)
    for (int idx = lane; idx < 16 * TILE_T; idx += 32) {
      int n = idx / TILE_T;
      int t = idx - n * TILE_T;
      int ia = a0 - RAD + t;
      ia = ia < 0 ? 0 : (ia >= La ? La - 1 : ia);
      async_ld_f32((uint32_t)(uintptr_t)&tile[n * TILE_PITCH + t],
                   in + base + ia + (l0 + n) * Sl);
    }
  } else {
    // Sl == 1: consecutive lanes walk the contiguous line axis (n-major)
    for (int idx = lane; idx < 16 * TILE_T; idx += 32) {
      int t = idx >> 4;
      int n = idx & 15;
      int ia = a0 - RAD + t;
      ia = ia < 0 ? 0 : (ia >= La ? La - 1 : ia);
      async_ld_f32((uint32_t)(uintptr_t)&tile[n * TILE_PITCH + t],
                   in + base + ia * Sa + (l0 + n));
    }
  }

  // zero-padded weight window: wpad[16+j] = w[j] for j in [0,25), else 0.
  wpad[lane]      = 0.0f;
  wpad[lane + 32] = 0.0f;
  if (lane < KW) wpad[WPAD_BASE + lane] = wAll[(b * 3 + waxis) * KW + lane];

  // Pull the next outer slice toward L2 while we compute (global_prefetch_b8).
  __builtin_prefetch(in + base + So + lane * 16, 0, 1);

  wait_async0();                              // async-copied tile visible
  __syncthreads();                            // DS ordering for wpad

  // ---- 10x chained WMMA; EXEC untouched in this region (WMMA requirement).
  //   A 16x4 f32: lanes 0-15 hold K={0,1}, lanes 16-31 K={2,3} (2 VGPRs)
  //   B 4x16 f32: row-striped, same K split across lane halves
  const int m  = lane & 15;                   // line index OR blur position
  const int kb = (lane < 16) ? 0 : 2;         // K-pair selector

  v8f acc = {};
#pragma unroll
  for (int kk = 0; kk < 10; ++kk) {
    const int t0 = kk * 4 + kb;               // even -> 8B-aligned ds_load_b64
    v2f fin = *(const v2f*)&tile[m * TILE_PITCH + t0];
    v2f fw;                                   // Toeplitz: T[p,t] = w[t-p], padded
    fw.x = wpad[WPAD_BASE + t0 - m];
    fw.y = wpad[WPAD_BASE + t0 + 1 - m];
    if (BLUR_ON_N) {
      // A = input lines, B = Toeplitz  (blur axis on N)
      acc = __builtin_amdgcn_wmma_f32_16x16x4_f32(
          false, fin, false, fw, (short)0, acc, false, false);
    } else {
      // A = Toeplitz, B = input lines  (blur axis on M)
      acc = __builtin_amdgcn_wmma_f32_16x16x4_f32(
          false, fw, false, fin, (short)0, acc, false, false);
    }
  }

  // ---- store D: row M = vgpr + (lane>=16 ? 8 : 0), col N = lane&15
  const int hi8 = (lane < 16) ? 0 : 8;
#pragma unroll
  for (int r = 0; r < 8; ++r) {
    int M = r + hi8;
    int ap = BLUR_ON_N ? m : M;               // position along blur axis
    int lp = BLUR_ON_N ? M : m;               // position along line axis
    out[base + (a0 + ap) * Sa + (l0 + lp) * Sl] = acc[r];
  }
}

// ---------------------------------------------------------------------------
// Harness entry: x:[4,2,144,144,144] f32, sigma:[4,3] f32 -> out f32 same shape
// ---------------------------------------------------------------------------
extern "C" void kernel_launch(void* const* d_in, const int* in_sizes, int n_in,
                              void* d_out, int out_size, void* d_ws, size_t ws_size,
                              hipStream_t stream) {
  const float* x     = (const float*)d_in[0];
  const float* sigma = (const float*)d_in[1];
  float* out  = (float*)d_out;
  float* wbuf = (float*)d_ws;                          // 12*25 floats of weights
  float* ping = (float*)((char*)d_ws + 4096);          // full-tensor ping buffer

  // geometry: [B=4, C=2, D=144, H=144, W=144]
  const int SH  = 144;
  const int SD  = 144 * 144;          // 20736
  const int SBC = 144 * 144 * 144;    // 2985984 (c stride; b stride = 2*SBC)
  const int tiles = 144 / 16;         // 9, exact

  gauss_weights_kernel<<<dim3(1), dim3(32), 0, stream>>>(sigma, wbuf);

  dim3 grid(tiles * tiles, 144, 8);   // (ta*tl, outer, b*c)
  dim3 blk(32);

  // Pass 1: blur W (waxis=2): Sa=1, lines=H, outer=D, x -> out
  blur_axis_wmma<1><<<grid, blk, 0, stream>>>(x, out, wbuf, 2,
                                              1, 144, SH, SD, SBC, tiles);
  // Pass 2: blur H (waxis=1): Sa=SH, lines=W, outer=D, out -> ping
  blur_axis_wmma<0><<<grid, blk, 0, stream>>>(out, ping, wbuf, 1,
                                              SH, 144, 1, SD, SBC, tiles);
  // Pass 3: blur D (waxis=0): Sa=SD, lines=W, outer=H, ping -> out
  blur_axis_wmma<0><<<grid, blk, 0, stream>>>(ping, out, wbuf, 0,
                                              SD, 144, 1, SH, SBC, tiles);
}